// FractalGenerator_30769145709115
// MI455X (gfx1250) — compile-verified
//
#include <hip/hip_runtime.h>

// out[b,r,c] = field[b,r,c] * (1 + 0.1 * (0.7*perlin(r,c) + 0.3*mandel(r,c)))
//
// Bandwidth-bound: 512 MB of HBM traffic (~22us @ 23.3 TB/s). Fractal field is
// input-independent and computed once per pixel, fused with the 16-batch
// streaming multiply. 128-bit NT loads/stores (no reuse, 256MB > 192MB L2),
// global_prefetch_b8 to overlap the mandelbrot loop with the first HBM fills.

typedef float f4 __attribute__((ext_vector_type(4)));

#define DIM     2048
#define HW_PIX  (DIM * DIM)
#define HW_VEC  (HW_PIX / 4)   // float4 per (h,w) slice

__global__ __launch_bounds__(256)
void fractal_apply_kernel(const float* __restrict__ field,
                          float* __restrict__ out,
                          int B) {
    const int tid = blockIdx.x * blockDim.x + threadIdx.x;
    if (tid >= HW_VEC) return;

    const f4* __restrict__ in4  = (const f4*)field;
    f4* __restrict__       out4 = (f4*)out;

    // Kick off HBM fills for the first batch slices; the fractal compute below
    // hides this latency. (gfx1250: global_prefetch_b8)
    __builtin_prefetch(&in4[tid], 0, 0);
    if (B > 1) __builtin_prefetch(&in4[(size_t)HW_VEC + tid], 0, 0);
    if (B > 2) __builtin_prefetch(&in4[(size_t)2 * HW_VEC + tid], 0, 0);

    // 512 float4 per row; 4 consecutive columns share one row.
    const int r  = tid >> 9;
    const int c0 = (tid & 511) << 2;
    const float inv = 1.0f / 2047.0f;   // linspace step denominators (N-1)

    // ---------------- Mandelbrot: 4 points, shared real coordinate ----------
    // rows -> x = linspace(-2.5, 1.5, 2048); cols -> y = linspace(-1.5, 1.5, 2048)
    const float cx = -2.5f + 4.0f * (float)r * inv;
    float cy[4], zx[4], zy[4], esc[4];
#pragma unroll
    for (int k = 0; k < 4; ++k) {
        cy[k]  = -1.5f + 3.0f * (float)(c0 + k) * inv;
        zx[k]  = 0.0f;
        zy[k]  = 0.0f;
        esc[k] = 0.0f;
    }
    for (int i = 0; i < 100; ++i) {
        bool any = false;
#pragma unroll
        for (int k = 0; k < 4; ++k) {
            const float m2 = zx[k] * zx[k] + zy[k] * zy[k];
            if (m2 <= 4.0f) {               // reference mask: |z| <= 2
                esc[k] = (float)i;          // esc = last inside iteration
                const float nx = zx[k] * zx[k] - zy[k] * zy[k] + cx;
                zy[k] = 2.0f * zx[k] * zy[k] + cy[k];
                zx[k] = nx;
                any = true;
            }
        }
        if (!any) break;                    // wave-level early exit (deterministic)
    }

    // ---------------- Perlin-ish 6-octave noise ----------------------------
    // rows -> X = linspace(0, 10, 2048); cols -> Y = linspace(0, 10, 2048)
    const float X = 10.0f * (float)r * inv;
    float noise[4] = {0.0f, 0.0f, 0.0f, 0.0f};
    float amp = 1.0f, freq = 1.0f;
#pragma unroll
    for (int o = 0; o < 6; ++o) {
        const float s = __sinf(freq * X);   // shared across the 4 columns
#pragma unroll
        for (int k = 0; k < 4; ++k) {
            const float Y = 10.0f * (float)(c0 + k) * inv;
            noise[k] += amp * s * __cosf(freq * Y);
        }
        amp  *= 0.5f;
        freq *= 2.0f;
    }

    // scale[k] = INTENSITY * (0.7*perlin + 0.3*mandel)
    f4 scale;
#pragma unroll
    for (int k = 0; k < 4; ++k) {
        const float perlin = (noise[k] + 1.0f) * 0.5f;
        const float mandel = esc[k] / 100.0f;
        scale[k] = 0.1f * (0.7f * perlin + 0.3f * mandel);
    }

    // ---------------- Stream the 16 batch slices ----------------------------
#pragma unroll 4
    for (int b = 0; b < B; ++b) {
        const size_t idx = (size_t)b * HW_VEC + tid;
        f4 v = __builtin_nontemporal_load(&in4[idx]);
        f4 w;
#pragma unroll
        for (int k = 0; k < 4; ++k)
            w[k] = fmaf(v[k], scale[k], v[k]);   // field + 0.1*fractal*field
        __builtin_nontemporal_store(w, &out4[idx]);
    }
}

extern "C" void kernel_launch(void* const* d_in, const int* in_sizes, int n_in,
                              void* d_out, int out_size, void* d_ws, size_t ws_size,
                              hipStream_t stream) {
    (void)n_in; (void)out_size; (void)d_ws; (void)ws_size;
    const float* field = (const float*)d_in[0];
    float* out = (float*)d_out;
    const int B = in_sizes[0] / HW_PIX;          // 16

    const int threads = 256;                      // 8 wave32 per block
    const int blocks  = (HW_VEC + threads - 1) / threads;  // 4096
    fractal_apply_kernel<<<blocks, threads, 0, stream>>>(field, out, B);
}